// Decoder_70918499992357
// MI455X (gfx1250) — compile-verified
//
#include <hip/hip_runtime.h>
#include <cstdint>

// ---------------------------------------------------------------------------
// Tacotron2-style decoder with stepwise monotonic attention for gfx1250.
// B=16, T_ENC=200, T_DEC=300, E=512, A_RNN=D_RNN=1024, P=256, A=128, N_MELS=80
// Strategy: weights (~73MB fp32) are L2-resident (192MB L2); sequential scan
// runs in ONE persistent kernel (32 WGs x 256 thr, wave32) with software grid
// barriers; all LSTM-gate GEMMs use V_WMMA_F32_16X16X4_F32 (exact fp32).
// ---------------------------------------------------------------------------

typedef float v2f __attribute__((ext_vector_type(2)));
typedef float v8f __attribute__((ext_vector_type(8)));

static constexpr int kTEnc = 200, kTDec = 300;
static constexpr int kNWG  = 32,  kNT   = 256;

// workspace layout (in floats)
static constexpr size_t WS_PRENET = 0;                               // 300*16*256
static constexpr size_t WS_PM     = WS_PRENET + (size_t)300*16*256;  // 16*200*128
static constexpr size_t WS_ATTH   = WS_PM     + (size_t)16*200*128;  // 2*16*1024
static constexpr size_t WS_DECH   = WS_ATTH   + (size_t)2*16*1024;   // 2*16*1024
static constexpr size_t WS_CTX    = WS_DECH   + (size_t)2*16*1024;   // 2*16*512
static constexpr size_t WS_ALGN   = WS_CTX    + (size_t)2*16*512;    // 2*16*200
static constexpr size_t WS_BAR    = WS_ALGN   + (size_t)2*16*200;    // 2 ints

// ---------------------------------------------------------------------------
// Threefry2x32 (JAX-compatible) for the fixed prenet dropout masks.
// ---------------------------------------------------------------------------
__host__ __device__ inline void threefry2x32(uint32_t k0, uint32_t k1,
                                             uint32_t c0, uint32_t c1,
                                             uint32_t& o0, uint32_t& o1) {
  const uint32_t ks2 = k0 ^ k1 ^ 0x1BD11BDAu;
  uint32_t x0 = c0 + k0, x1 = c1 + k1;
#define TF_R(r) { x0 += x1; x1 = (x1 << (r)) | (x1 >> (32 - (r))); x1 ^= x0; }
  TF_R(13) TF_R(15) TF_R(26) TF_R(6)  x0 += k1;  x1 += ks2 + 1u;
  TF_R(17) TF_R(29) TF_R(16) TF_R(24) x0 += ks2; x1 += k0 + 2u;
  TF_R(13) TF_R(15) TF_R(26) TF_R(6)  x0 += k0;  x1 += k1 + 3u;
  TF_R(17) TF_R(29) TF_R(16) TF_R(24) x0 += k1;  x1 += ks2 + 4u;
  TF_R(13) TF_R(15) TF_R(26) TF_R(6)  x0 += ks2; x1 += k0 + 5u;
#undef TF_R
  o0 = x0; o1 = x1;
}

__device__ inline float dropout_mult(uint32_t k0, uint32_t k1, uint32_t idx, uint32_t n) {
  const uint32_t half = n >> 1;
  const uint32_t c0 = idx < half ? idx : idx - half;
  const uint32_t c1 = idx < half ? idx + half : idx;
  uint32_t o0, o1; threefry2x32(k0, k1, c0, c1, o0, o1);
  const uint32_t bits = idx < half ? o0 : o1;
  const float u = __uint_as_float((bits >> 9) | 0x3f800000u) - 1.0f;  // [0,1)
  return u < 0.5f ? 2.0f : 0.0f;  // keep-with-prob-0.5, scale 1/p
}

__device__ inline float sigmoidf(float x) { return 1.0f / (1.0f + expf(-x)); }

// ---------------------------------------------------------------------------
// Wave-level 16x16 fp32 GEMM span: acc += X(16xkn) * W^T(knx16) via
// V_WMMA_F32_16X16X4_F32.  A frag: lane m=lane&15, VGPR pair = K(2*kh),(2*kh+1)
// with kh=lane>>4 (ISA 7.12.2 32-bit A 16x4).  B frag symmetric (K rows striped
// across the two VGPRs, N=lane&15).  x = row-m region base, w = per-column row.
// ---------------------------------------------------------------------------
__device__ __forceinline__ v8f wmma_span(v8f acc, const float* __restrict__ x,
                                         const float* __restrict__ w,
                                         int kn, int kh2) {
#pragma unroll 4
  for (int k = 0; k < kn; k += 4) {
    v2f a = *(const v2f*)(x + k + kh2);
    v2f b = *(const v2f*)(w + k + kh2);
    acc = __builtin_amdgcn_wmma_f32_16x16x4_f32(false, a, false, b,
                                                (short)0, acc, false, false);
  }
  return acc;
}

// Stage the 16x16 gate tiles into LDS and run the LSTM pointwise for this WG's
// 32 units.  Cell state stays in registers (creg) across all 300 steps.
__device__ __forceinline__ void lstm_tail(float* gb, const v8f& acc,
                                          int g, int s, int n, int lane, int tid, int u0,
                                          const float* __restrict__ b_ih,
                                          const float* __restrict__ b_hh,
                                          float* hout, float* creg) {
#pragma unroll
  for (int r = 0; r < 8; ++r)  // C layout: M = r + 8*(lane>>4), N = lane&15
    gb[(g * 32 + s * 16 + n) * 16 + r + ((lane >> 4) << 3)] = acc[r];
  __syncthreads();
  const int b = tid & 15, j = tid >> 4;
#pragma unroll
  for (int q = 0; q < 2; ++q) {
    const int uu = j + q * 16, u = u0 + uu;
    const float iv = gb[(0 * 32 + uu) * 16 + b] + b_ih[u]          + b_hh[u];
    const float fv = gb[(1 * 32 + uu) * 16 + b] + b_ih[1024 + u]   + b_hh[1024 + u];
    const float gv = gb[(2 * 32 + uu) * 16 + b] + b_ih[2048 + u]   + b_hh[2048 + u];
    const float ov = gb[(3 * 32 + uu) * 16 + b] + b_ih[3072 + u]   + b_hh[3072 + u];
    const float c2 = sigmoidf(fv) * creg[q] + sigmoidf(iv) * tanhf(gv);
    creg[q] = c2;
    hout[b * 1024 + u] = sigmoidf(ov) * tanhf(c2);
  }
  __syncthreads();
}

// Software grid barrier (counter + generation), re-zeroed each launch.
__device__ __forceinline__ void grid_sync(int* cnt, int* gen, int nwg, int& epoch) {
  __syncthreads();
  if (threadIdx.x == 0) {
    __threadfence();
    epoch += 1;
    const int prev = atomicAdd(cnt, 1);
    if (prev == nwg - 1) {
      atomicExch(cnt, 0);
      __threadfence();
      atomicAdd(gen, 1);
    } else {
      while (atomicAdd(gen, 0) < epoch) __builtin_amdgcn_s_sleep(2);
    }
    __threadfence();
  }
  __syncthreads();
}

// ---------------------------------------------------------------------------
// Pre-pass kernels
// ---------------------------------------------------------------------------
__global__ void init_state_kernel(float* state /* att_h..ctx */, float* algn, int* bar) {
  const int i = blockIdx.x * blockDim.x + threadIdx.x;
  const int stride = gridDim.x * blockDim.x;
  for (int k = i; k < 2 * 16 * 1024 + 2 * 16 * 1024 + 2 * 16 * 512; k += stride)
    state[k] = 0.f;
  for (int k = i; k < 2 * 16 * 200; k += stride)
    algn[k] = ((k % 200) == 0) ? 1.f : 0.f;   // SMA init: align[:,0] = 1
  if (i == 0) { bar[0] = 0; bar[1] = 0; }
}

// Prenet: go-frame shift + two bias-free linears + relu + fixed dropout masks.
__global__ void prenet_kernel(const float* __restrict__ mels,
                              const float* __restrict__ w1,
                              const float* __restrict__ w2,
                              float* __restrict__ prenet_out,
                              uint32_t k1a, uint32_t k1b, uint32_t k2a, uint32_t k2b) {
  __shared__ float x1s[256];
  const int blk = blockIdx.x;          // t*16 + b
  const int t = blk >> 4, b = blk & 15;
  const int j = threadIdx.x;
  const uint32_t N = 300u * 16u * 256u;
  float s = 0.f;
  if (t > 0) {
    const float* mrow = mels + (size_t)b * 80 * 300 + (t - 1);  // stride 300 per mel bin
    const float* wr = w1 + (size_t)j * 80;
    for (int i = 0; i < 80; ++i) s += mrow[(size_t)i * 300] * wr[i];
  }
  s = fmaxf(s, 0.f);
  const uint32_t idx = (uint32_t)blk * 256u + (uint32_t)j;
  s *= dropout_mult(k1a, k1b, idx, N);
  x1s[j] = s;
  __syncthreads();
  const float* wr2 = w2 + (size_t)j * 256;
  float s2 = 0.f;
  for (int p = 0; p < 256; ++p) s2 += x1s[p] * wr2[p];
  s2 = fmaxf(s2, 0.f);
  s2 *= dropout_mult(k2a, k2b, idx, N);
  prenet_out[(size_t)blk * 256 + j] = s2;
}

// processed_memory[b,t,a] = sum_e memory[b,t,e] * mem_w[a,e]
__global__ void pm_kernel(const float* __restrict__ memory,
                          const float* __restrict__ mem_w,
                          float* __restrict__ pm) {
  const int bt = blockIdx.x;           // b*200 + t
  const int a = threadIdx.x;           // 128 threads
  const float* mrow = memory + (size_t)bt * 512;
  const float* wr = mem_w + (size_t)a * 512;
  float s = 0.f;
  for (int e = 0; e < 512; ++e) s += mrow[e] * wr[e];
  pm[(size_t)bt * 128 + a] = s;
}

// ---------------------------------------------------------------------------
// Persistent sequential decoder: 32 WGs, 3 grid barriers per step.
// ---------------------------------------------------------------------------
__global__ void __launch_bounds__(256, 1)
tacotron_decoder_persistent(
    const float* __restrict__ memory, const float* __restrict__ prenet_out,
    const float* __restrict__ pmem,
    const float* __restrict__ query_w, const float* __restrict__ v_w,
    const float* __restrict__ att_w_ih, const float* __restrict__ att_w_hh,
    const float* __restrict__ att_b_ih, const float* __restrict__ att_b_hh,
    const float* __restrict__ dec_w_ih, const float* __restrict__ dec_w_hh,
    const float* __restrict__ dec_b_ih, const float* __restrict__ dec_b_hh,
    const float* __restrict__ proj_w, const float* __restrict__ proj_b,
    const float* __restrict__ gate_w, const float* __restrict__ gate_b,
    float* att_h, float* dec_h, float* ctx, float* algn, int* bar,
    float* out_mel, float* out_gate, float* out_align) {
  __shared__ union {
    float gbuf[4 * 32 * 16];                               // [gate][unit][batch]
    struct { float pq[128]; float p[224]; float aln[224]; } att;
  } sh;

  const int tid  = threadIdx.x;
  const int wg   = blockIdx.x;
  const int lane = tid & 31;
  const int wave = tid >> 5;
  const int g = wave >> 1;                  // gate 0..3 (i,f,g,o)
  const int s = wave & 1;                   // which 16-unit half of the chunk
  const int n = lane & 15;                  // tile column
  const int m = lane & 15;                  // tile row (batch)
  const int kh2 = (lane >> 4) << 1;         // 0 or 2: K-pair select for frags
  const int u0 = wg * 32;                   // this WG's unit chunk
  const int col = g * 1024 + u0 + s * 16 + n;  // gate-matrix output column

  float cA[2] = {0.f, 0.f};                 // attention-LSTM cell state (regs)
  float cD[2] = {0.f, 0.f};                 // decoder-LSTM cell state (regs)
  int epoch = 0;

  for (int t = 0; t <= kTDec; ++t) {        // extra iter flushes last projection
    const int cur = t & 1, prv = cur ^ 1;

    // ---- phase A: attention LSTM (all 32 WGs) ----
    if (t < kTDec) {
      const float* xr0 = prenet_out + ((size_t)t * 16 + m) * 256;   // xt
      const float* xr1 = ctx   + prv * (16 * 512)  + m * 512;       // ctx_{t-1}
      const float* xr2 = att_h + prv * (16 * 1024) + m * 1024;      // h_{t-1}
      const float* wih = att_w_ih + (size_t)col * 768;
      const float* whh = att_w_hh + (size_t)col * 1024;
      v8f acc = {};
      acc = wmma_span(acc, xr0, wih,       256,  kh2);
      acc = wmma_span(acc, xr1, wih + 256, 512,  kh2);
      acc = wmma_span(acc, xr2, whh,       1024, kh2);
      lstm_tail(sh.gbuf, acc, g, s, n, lane, tid, u0, att_b_ih, att_b_hh,
                att_h + cur * (16 * 1024), cA);
    }
    grid_sync(bar, bar + 1, kNWG, epoch);

    // ---- phase B: attention (WGs 0..15, one batch row each) ----
    if (t < kTDec && wg < 16) {
      const int b = wg;
      const float* hcur = att_h + cur * (16 * 1024) + b * 1024;
      if (tid < 128) {                                  // pq = h @ query_w.T
        const float* qw = query_w + (size_t)tid * 1024;
        float sum = 0.f;
        for (int k = 0; k < 1024; ++k) sum += hcur[k] * qw[k];
        sh.att.pq[tid] = sum;
      }
      __syncthreads();
      if (tid < 200) {                                  // p = sigmoid(v.tanh(pq+pm))
        const float* pmr = pmem + ((size_t)b * 200 + tid) * 128;
        float e = 0.f;
        for (int a2 = 0; a2 < 128; ++a2) e += v_w[a2] * tanhf(sh.att.pq[a2] + pmr[a2]);
        sh.att.p[tid] = sigmoidf(e);
      }
      __syncthreads();
      const float* alo = algn + prv * (16 * 200) + b * 200;
      float* alc = algn + cur * (16 * 200) + b * 200;
      if (tid < 200) {                                  // SMA recurrence
        float v = alo[tid] * sh.att.p[tid];
        if (tid > 0) v += alo[tid - 1] * (1.f - sh.att.p[tid - 1]);
        sh.att.aln[tid] = v;
        alc[tid] = v;
        out_align[(size_t)b * kTDec * kTEnc + (size_t)t * kTEnc + tid] = v;
      }
      __syncthreads();
      float* cxc = ctx + cur * (16 * 512) + b * 512;    // ctx = align @ memory
      for (int e0 = tid; e0 < 512; e0 += 256) {
        const float* mb = memory + (size_t)b * 200 * 512 + e0;
        float sum = 0.f;
        for (int tt = 0; tt < 200; ++tt) sum += sh.att.aln[tt] * mb[(size_t)tt * 512];
        cxc[e0] = sum;
      }
    }
    // ---- phase B (overlapped): projections for step t-1 (WGs 16..31) ----
    if (t >= 1 && wg >= 16) {
      const int b = wg - 16;
      const float* dh = dec_h + prv * (16 * 1024) + b * 1024;  // (t-1)&1 == prv
      const float* cx = ctx   + prv * (16 * 512)  + b * 512;
      if (tid < 80) {
        const float* pw = proj_w + (size_t)tid * 1536;
        float sum = proj_b[tid];
        for (int k = 0; k < 1024; ++k) sum += dh[k] * pw[k];
        for (int k = 0; k < 512;  ++k) sum += cx[k] * pw[1024 + k];
        out_mel[(size_t)b * 80 * kTDec + (size_t)tid * kTDec + (t - 1)] = sum;
      } else if (tid == 80) {
        float sum = gate_b[0];
        for (int k = 0; k < 1024; ++k) sum += dh[k] * gate_w[k];
        for (int k = 0; k < 512;  ++k) sum += cx[k] * gate_w[1024 + k];
        out_gate[(size_t)b * kTDec + (t - 1)] = sum;
      }
    }
    grid_sync(bar, bar + 1, kNWG, epoch);

    // ---- phase C: decoder LSTM (all 32 WGs) ----
    if (t < kTDec) {
      const float* xr0 = att_h + cur * (16 * 1024) + m * 1024;  // att_h_t
      const float* xr1 = ctx   + cur * (16 * 512)  + m * 512;   // ctx_t
      const float* xr2 = dec_h + prv * (16 * 1024) + m * 1024;  // dec_h_{t-1}
      const float* wih = dec_w_ih + (size_t)col * 1536;
      const float* whh = dec_w_hh + (size_t)col * 1024;
      v8f acc = {};
      acc = wmma_span(acc, xr0, wih,        1024, kh2);
      acc = wmma_span(acc, xr1, wih + 1024, 512,  kh2);
      acc = wmma_span(acc, xr2, whh,        1024, kh2);
      lstm_tail(sh.gbuf, acc, g, s, n, lane, tid, u0, dec_b_ih, dec_b_hh,
                dec_h + cur * (16 * 1024), cD);
    }
    grid_sync(bar, bar + 1, kNWG, epoch);
  }
}

// ---------------------------------------------------------------------------
extern "C" void kernel_launch(void* const* d_in, const int* in_sizes, int n_in,
                              void* d_out, int out_size, void* d_ws, size_t ws_size,
                              hipStream_t stream) {
  (void)in_sizes; (void)n_in; (void)out_size; (void)ws_size;
  const float* memory   = (const float*)d_in[0];
  const float* mels     = (const float*)d_in[1];
  const float* pw1      = (const float*)d_in[2];
  const float* pw2      = (const float*)d_in[3];
  const float* mem_w    = (const float*)d_in[4];
  const float* query_w  = (const float*)d_in[5];
  const float* v_w      = (const float*)d_in[6];
  const float* att_w_ih = (const float*)d_in[7];
  const float* att_w_hh = (const float*)d_in[8];
  const float* att_b_ih = (const float*)d_in[9];
  const float* att_b_hh = (const float*)d_in[10];
  const float* dec_w_ih = (const float*)d_in[11];
  const float* dec_w_hh = (const float*)d_in[12];
  const float* dec_b_ih = (const float*)d_in[13];
  const float* dec_b_hh = (const float*)d_in[14];
  const float* proj_w   = (const float*)d_in[15];
  const float* proj_b   = (const float*)d_in[16];
  const float* gate_w   = (const float*)d_in[17];
  const float* gate_b   = (const float*)d_in[18];

  float* ws     = (float*)d_ws;
  float* prenet = ws + WS_PRENET;
  float* pmem   = ws + WS_PM;
  float* att_h  = ws + WS_ATTH;
  float* dec_h  = ws + WS_DECH;
  float* ctx    = ws + WS_CTX;
  float* algn   = ws + WS_ALGN;
  int*   bar    = (int*)(ws + WS_BAR);

  float* out       = (float*)d_out;
  float* out_mel   = out;                                  // (16,80,300)
  float* out_gate  = out + (size_t)16 * 80 * 300;          // (16,300)
  float* out_align = out_gate + (size_t)16 * 300;          // (16,300,200)

  // kd1, kd2 = jax.random.split(jax.random.key(42)); key(42) = (0,42)
  uint32_t a0, b0, a1, b1;
  threefry2x32(0u, 42u, 0u, 2u, a0, b0);
  threefry2x32(0u, 42u, 1u, 3u, a1, b1);
  // kd1 = (a0, a1), kd2 = (b0, b1)

  hipLaunchKernelGGL(init_state_kernel, dim3(128), dim3(256), 0, stream,
                     att_h, algn, bar);
  hipLaunchKernelGGL(prenet_kernel, dim3(300 * 16), dim3(256), 0, stream,
                     mels, pw1, pw2, prenet, a0, a1, b0, b1);
  hipLaunchKernelGGL(pm_kernel, dim3(16 * 200), dim3(128), 0, stream,
                     memory, mem_w, pmem);
  hipLaunchKernelGGL(tacotron_decoder_persistent, dim3(kNWG), dim3(kNT), 0, stream,
                     memory, prenet, pmem, query_w, v_w,
                     att_w_ih, att_w_hh, att_b_ih, att_b_hh,
                     dec_w_ih, dec_w_hh, dec_b_ih, dec_b_hh,
                     proj_w, proj_b, gate_w, gate_b,
                     att_h, dec_h, ctx, algn, bar,
                     out_mel, out_gate, out_align);
}